// VariationalQuantumLayer_65481071408688
// MI455X (gfx1250) — compile-verified
//
#include <hip/hip_runtime.h>
#include <hip/hip_bf16.h>

// out[row, e] = cos(x[row, 1]) * W_dec[e] + b_dec[e]
// rows = 8*4096 = 32768, e = 1024. Output 128 MiB f32 -> pure store-BW bound
// (~5.6 us at 23.3 TB/s). Rank-1 update mapped onto V_WMMA_F32_16X16X4_F32:
//   A (16x4, M=e, only K=0): W[e0+M]
//   B (4x16, K, N=row, only K=0): cos(x[row,1])
//   C (16x16): b[e0+M] broadcast over N
// D layout gives each lane 8 *contiguous* e-values -> two global_store_b128.
// This revision software-pipelines the 8 x-loads into one clause and turns
// all per-tile addressing into constant immediate offsets (tile stride
// 65536 B < 2^23) from two base pointers.

typedef __attribute__((ext_vector_type(2))) float v2f;
typedef __attribute__((ext_vector_type(4))) float f4;
typedef __attribute__((ext_vector_type(8))) float v8f;

#define EMBED 1024
#define ROWS (8 * 4096)
#define ROW_TILES_PER_BLOCK 8   // 8 tiles * 16 rows = 128 rows per block

__global__ __launch_bounds__(256) void vql_wmma_kernel(
    const float* __restrict__ x,
    const float* __restrict__ W,
    const float* __restrict__ bvec,
    float* __restrict__ out) {
  const int lane = threadIdx.x & 31;   // wave32 on gfx1250
  const int wave = threadIdx.x >> 5;   // 8 waves per block
  const int hi   = lane >> 4;          // 0: lanes 0-15, 1: lanes 16-31
  const int lo   = lane & 15;

  const int e0       = blockIdx.y * 128 + wave * 16;  // 16-wide e-tile per wave
  const int row_base = blockIdx.x * (ROW_TILES_PER_BLOCK * 16);

  // A: 16x4 f32 (M=e, K). VGPR0 holds K=0 (lanes 0-15) / K=2 (lanes 16-31);
  // VGPR1 holds K=1 / K=3. Only K=0 is non-zero: W[e0 + M].
  v2f A;
  A.x = (hi == 0) ? W[e0 + lo] : 0.0f;
  A.y = 0.0f;

  // C: 16x16 f32 (M=e offset, N=row). Element (M,N) lives in VGPR r = M&7,
  // lane = (M>=8 ? 16:0) + N. So per lane: C[r] = b[e0 + r + 8*hi], same
  // for every N (bias broadcast along rows).
  v8f C;
  {
    const float* bp = bvec + e0 + 8 * hi;
#pragma unroll
    for (int r = 0; r < 8; ++r) C[r] = bp[r];
  }

  // Base pointers; all per-tile addressing becomes immediate offsets
  // (t * 16 * EMBED * 4 = t * 65536 bytes, within signed 24-bit ioffset).
  const float* xp = x + (size_t)(row_base + lo) * EMBED + 1;
  float* op = out + (size_t)(row_base + lo) * EMBED + e0 + 8 * hi;

  const float INV_2PI = 0.15915494309189535f;  // v_cos_f32 input in revolutions

  // Stage 1: issue all 8 independent x-loads as one clause.
  float cv[ROW_TILES_PER_BLOCK];
#pragma unroll
  for (int t = 0; t < ROW_TILES_PER_BLOCK; ++t)
    cv[t] = xp[(size_t)t * 16 * EMBED];

  // Stage 2: cos + lane-mask (B matrix K=0 slot lives in lanes 0-15 only).
#pragma unroll
  for (int t = 0; t < ROW_TILES_PER_BLOCK; ++t) {
    const float c = __builtin_amdgcn_cosf(cv[t] * INV_2PI);
    cv[t] = (hi == 0) ? c : 0.0f;
  }

  // Stage 3: 8x (wmma + two b128 stores), no memory stalls in between.
#pragma unroll
  for (int t = 0; t < ROW_TILES_PER_BLOCK; ++t) {
    v2f B;
    B.x = cv[t];
    B.y = 0.0f;

    v8f D = __builtin_amdgcn_wmma_f32_16x16x4_f32(
        /*neg_a=*/false, A, /*neg_b=*/false, B,
        /*c_mod=*/(short)0, C, /*reuse_a=*/false, /*reuse_b=*/false);

    float* p = op + (size_t)t * 16 * EMBED;
    f4 d_lo = {D[0], D[1], D[2], D[3]};
    f4 d_hi = {D[4], D[5], D[6], D[7]};
    *reinterpret_cast<f4*>(p)     = d_lo;
    *reinterpret_cast<f4*>(p + 4) = d_hi;
  }
}

extern "C" void kernel_launch(void* const* d_in, const int* in_sizes, int n_in,
                              void* d_out, int out_size, void* d_ws, size_t ws_size,
                              hipStream_t stream) {
  const float* x    = (const float*)d_in[0];  // (8, 4096, 1024) f32
  const float* Wdec = (const float*)d_in[1];  // (1024, 1) f32
  const float* bdec = (const float*)d_in[2];  // (1024,) f32
  float*       out  = (float*)d_out;          // (8, 4096, 1024) f32

  (void)in_sizes; (void)n_in; (void)out_size; (void)d_ws; (void)ws_size;

  dim3 grid(ROWS / (ROW_TILES_PER_BLOCK * 16), EMBED / 128, 1);  // (256, 8)
  dim3 block(256, 1, 1);  // 8 waves; each wave owns one 16-wide e-tile
  vql_wmma_kernel<<<grid, block, 0, stream>>>(x, Wdec, bdec, out);
}